// NodeConv_73650099192496
// MI455X (gfx1250) — compile-verified
//
#include <hip/hip_runtime.h>

#define N_NODES 100000
#define N_EDGES 1600000
#define D 64
#define TILES (N_NODES / 16)   // 6250, exact

typedef __attribute__((ext_vector_type(2))) float v2f;
typedef __attribute__((ext_vector_type(8))) float v8f;

// ---------------------------------------------------------------------------
// Kernel 1: zero the aggregation buffer (workspace)
// ---------------------------------------------------------------------------
__global__ __launch_bounds__(256) void zero_agg_kernel(float4* __restrict__ agg4, int n4) {
    int i = blockIdx.x * 256 + threadIdx.x;
    if (i < n4) agg4[i] = make_float4(0.f, 0.f, 0.f, 0.f);
}

// ---------------------------------------------------------------------------
// Kernel 2: edge scatter-add.  One wave32 per edge; each lane handles two
// consecutive features -> fully coalesced 256B gather of x[row], then two
// f32 atomic adds into agg[col].  Result of the atomic is unused so the
// backend selects the no-return GLOBAL_ATOMIC_ADD_F32.
// All of x and agg fit in the 192MB L2, so these atomics stay on-chip.
// ---------------------------------------------------------------------------
__global__ __launch_bounds__(256) void edge_scatter_kernel(
    const float* __restrict__ x,
    const int*   __restrict__ row,
    const int*   __restrict__ col,
    float*       __restrict__ agg) {
    int tid  = threadIdx.x;
    int e    = blockIdx.x * 8 + (tid >> 5);   // N_EDGES % 8 == 0, no guard
    int lane = tid & 31;
    int r = row[e];
    int c = col[e];
    int f = lane * 2;
    const float2 v = *(const float2*)(x + (long)r * D + f);
    float* dst = agg + (long)c * D + f;
    (void)unsafeAtomicAdd(dst,     v.x);
    (void)unsafeAtomicAdd(dst + 1, v.y);
}

// ---------------------------------------------------------------------------
// Kernel 3: fused  relu( (agg/adj_norm) @ rel_W^T  +  x @ root_W^T + root_b )
//
// Per workgroup: stage both 64x64 f32 weight matrices into LDS in WMMA
// B-fragment order:  lds[mat][nt][ks][lane] = W[j,k0..k0+1]  with
//   j = nt*16 + (lane&15),  k0 = ks*4 + (lane>>4)*2      (B mirrors A layout)
// Per wave: one 16-node tile, 4 N-tiles of v8f accumulators seeded with the
// bias, K=64 consumed as 16 x V_WMMA_F32_16X16X4_F32 per (matrix, N-tile).
// ---------------------------------------------------------------------------
__global__ __launch_bounds__(256) void fused_gemm_kernel(
    const float* __restrict__ x,
    const float* __restrict__ agg,
    const float* __restrict__ adj_norm,
    const float* __restrict__ root_W,
    const float* __restrict__ root_b,
    const float* __restrict__ rel_W,
    float*       __restrict__ out) {
    __shared__ v2f lds[4096];   // 2 mats * 4 nt * 16 ks * 32 lanes = 32 KB

    int tid = threadIdx.x;

    // ---- stage weights into LDS (B-fragment layout) ----
    for (int e = tid; e < 4096; e += 256) {
        int lane = e & 31;
        int ks   = (e >> 5) & 15;
        int nt   = (e >> 9) & 3;
        int mat  = e >> 11;                 // 0 = rel_W (for agg), 1 = root_W (for x)
        int h    = lane >> 4;
        int j    = nt * 16 + (lane & 15);   // output feature (N)
        int k    = ks * 4 + h * 2;          // reduction index (K)
        const float* W = (mat == 0) ? rel_W : root_W;
        float2 w = *(const float2*)(W + j * D + k);
        v2f wv;  wv.x = w.x;  wv.y = w.y;
        lds[e] = wv;
    }
    __syncthreads();

    int wave = tid >> 5;
    int lane = tid & 31;
    int tile = blockIdx.x * 8 + wave;       // wave-uniform guard: EXEC stays all-1s
    if (tile >= TILES) return;

    int m      = lane & 15;                 // M within tile (A/C row, C col index)
    int h      = lane >> 4;
    int node_m = tile * 16 + m;

    float scale = 1.0f / adj_norm[node_m];  // agg / adj_norm

    // ---- accumulators seeded with bias: C[v] = root_b[nt*16 + N], N = lane&15 ----
    v8f acc[4];
#pragma unroll
    for (int nt = 0; nt < 4; ++nt) {
        float b = root_b[nt * 16 + m];
        acc[nt] = (v8f){b, b, b, b, b, b, b, b};
    }

    const float* aggp = agg + (long)node_m * D;
    const float* xp   = x   + (long)node_m * D;

    // ---- K loop: 16 steps of K=4, two matrices, 4 N-tiles each ----
#pragma unroll
    for (int ks = 0; ks < 16; ++ks) {
        int k0 = ks * 4 + h * 2;            // A layout: lanes 0-15 K=0,1; 16-31 K=2,3
        float2 a2 = *(const float2*)(aggp + k0);
        v2f a_agg;  a_agg.x = a2.x * scale;  a_agg.y = a2.y * scale;
        float2 x2 = *(const float2*)(xp + k0);
        v2f a_x;    a_x.x = x2.x;            a_x.y = x2.y;
#pragma unroll
        for (int nt = 0; nt < 4; ++nt) {
            v2f b_rel  = lds[(nt * 16 + ks) * 32 + lane];
            acc[nt] = __builtin_amdgcn_wmma_f32_16x16x4_f32(
                false, a_agg, false, b_rel, (short)0, acc[nt], false, false);
            v2f b_root = lds[((4 + nt) * 16 + ks) * 32 + lane];
            acc[nt] = __builtin_amdgcn_wmma_f32_16x16x4_f32(
                false, a_x, false, b_root, (short)0, acc[nt], false, false);
        }
    }

    // ---- ReLU + store.  C layout: VGPR v -> M = v + 8*h, N = lane&15 ----
#pragma unroll
    for (int nt = 0; nt < 4; ++nt) {
#pragma unroll
        for (int v = 0; v < 8; ++v) {
            int node  = tile * 16 + h * 8 + v;
            float val = acc[nt][v];
            val = val > 0.f ? val : 0.f;
            out[(long)node * D + nt * 16 + m] = val;
        }
    }
}

// ---------------------------------------------------------------------------
// Host-side launcher
// ---------------------------------------------------------------------------
extern "C" void kernel_launch(void* const* d_in, const int* in_sizes, int n_in,
                              void* d_out, int out_size, void* d_ws, size_t ws_size,
                              hipStream_t stream) {
    const float* x        = (const float*)d_in[0];
    const int*   row      = (const int*)  d_in[1];
    const int*   col      = (const int*)  d_in[2];
    /* d_in[3] = batch, unused */
    const float* adj_norm = (const float*)d_in[4];
    const float* root_W   = (const float*)d_in[5];
    const float* root_b   = (const float*)d_in[6];
    const float* rel_W    = (const float*)d_in[7];
    float*       out      = (float*)d_out;
    float*       agg      = (float*)d_ws;       // N_NODES*D floats = 25.6 MB

    (void)in_sizes; (void)n_in; (void)out_size; (void)ws_size;

    int n4 = N_NODES * D / 4;
    zero_agg_kernel<<<(n4 + 255) / 256, 256, 0, stream>>>((float4*)agg, n4);

    edge_scatter_kernel<<<N_EDGES / 8, 256, 0, stream>>>(x, row, col, agg);

    int nblocks = (TILES + 7) / 8;              // 8 waves (tiles) per block
    fused_gemm_kernel<<<nblocks, 256, 0, stream>>>(x, agg, adj_norm,
                                                   root_W, root_b, rel_W, out);
}